// AnchorTargetLayer_48052094107725
// MI455X (gfx1250) — compile-verified
//
#include <hip/hip_runtime.h>
#include <hip/hip_bf16.h>
#include <stdint.h>

#define NBATCH 4
#define NGT    50
#define KANCH  240000
#define NUM_SEL 256u         // NUM_FG == NUM_BG == 256
#define NEG_INF_F (-1.0e9f)
#define NBUCK  65536         // 16-bit coarse buckets
#define NPROB  (NBATCH * 2)  // (batch, pos|neg)
#define LISTCAP 32768        // capacity of boundary-bucket tie list

typedef __attribute__((ext_vector_type(4))) unsigned int v4u;
typedef __attribute__((ext_vector_type(8))) int          v8i;
typedef __attribute__((ext_vector_type(4))) int          v4i;

// Map float to monotonically increasing unsigned key (IEEE total order).
__device__ __forceinline__ unsigned int orderKey(float f) {
    unsigned int b = __float_as_uint(f);
    return (b & 0x80000000u) ? ~b : (b | 0x80000000u);
}

__device__ __forceinline__ unsigned long long compKey(float key, int k) {
    // tie-break: lower index -> larger composite (jax top_k picks lowest index)
    return ((unsigned long long)orderKey(key) << 32) |
           (unsigned long long)(0xFFFFFFFFu - (unsigned int)k);
}

// ---------------------------------------------------------------------------
// Kernel A: per-anchor IoU max/argmax against 50 GT boxes (staged into LDS via
// the CDNA5 Tensor Data Mover), pos/neg flags, cls_targets, reg_targets.
// ---------------------------------------------------------------------------
__global__ void anchor_target_kernel(const float* __restrict__ anchors,
                                     const float* __restrict__ gt_boxes,
                                     const int*   __restrict__ gt_labels,
                                     float*       __restrict__ cls_t,
                                     float*       __restrict__ reg_t,
                                     unsigned char* __restrict__ flags) {
    __shared__ float4 s_gt[NGT];
    __shared__ int    s_lab[NGT];
    const int n = blockIdx.y;

#if __has_builtin(__builtin_amdgcn_tensor_load_to_lds)
    if (threadIdx.x == 0) {
        // --- Tensor DMA descriptor (D#) for an 800-byte 1-row tile ---
        unsigned long long ga =
            (unsigned long long)(uintptr_t)(gt_boxes + (size_t)n * NGT * 4);
        unsigned int lds = (unsigned int)(uintptr_t)(void*)&s_gt[0]; // addr[31:0] == LDS offset
        v4u g0;
        g0[0] = 1u;                                        // count = 1 (valid descriptor)
        g0[1] = lds;                                       // lds_addr
        g0[2] = (unsigned int)ga;                          // global_addr[31:0]
        g0[3] = (unsigned int)((ga >> 32) & 0x1FFFFFFu)    // global_addr[56:32]
              | (2u << 30);                                // type = 2 ("image")
        v8i g1;
        g1[0] = 0x00020000;            // data_size = 2 -> 4 bytes/element
        g1[1] = (int)(200u << 16);     // tensor_dim0[15:0] = 200 elements
        g1[2] = (int)(1u   << 16);     // tensor_dim1[15:0] = 1
        g1[3] = (int)(200u << 16);     // tile_dim0 = 200
        g1[4] = 1;                     // tile_dim1 = 1
        g1[5] = 200;                   // tensor_dim0_stride (low 32)
        g1[6] = 0;
        g1[7] = 0;
        v4i z4 = {0, 0, 0, 0};
        v8i z8 = {0, 0, 0, 0, 0, 0, 0, 0};
        __builtin_amdgcn_tensor_load_to_lds(g0, g1, z4, z4, z8, 0);
        __builtin_amdgcn_s_wait_tensorcnt(0);              // TENSORcnt == 0
    }
#else
    if (threadIdx.x < NGT) {
        const float4* src = (const float4*)(gt_boxes + (size_t)n * NGT * 4);
        s_gt[threadIdx.x] = src[threadIdx.x];
    }
#endif
    if (threadIdx.x < NGT)
        s_lab[threadIdx.x] = gt_labels[n * NGT + threadIdx.x];
    __syncthreads();

    const int k = blockIdx.x * blockDim.x + threadIdx.x;
    if (k >= KANCH) return;

    const float4 a = ((const float4*)anchors)[k];
    const float area_a = (a.z - a.x) * (a.w - a.y);

    float best = -1.0f;
    int   bm   = 0;
#pragma unroll 5
    for (int m = 0; m < NGT; ++m) {
        const float4 g = s_gt[m];
        const float lx = fmaxf(a.x, g.x), ly = fmaxf(a.y, g.y);
        const float rx = fminf(a.z, g.z), ry = fminf(a.w, g.w);
        const float w = fmaxf(rx - lx, 0.0f), h = fmaxf(ry - ly, 0.0f);
        const float inter  = w * h;
        const float area_g = (g.z - g.x) * (g.w - g.y);
        const float iou = inter / (area_a + area_g - inter + 1e-8f);
        if (iou > best) { best = iou; bm = m; }   // strict '>' keeps first max
    }

    const bool pos = best >= 0.7f;
    const bool neg = best <  0.3f;
    const size_t idx = (size_t)n * KANCH + (size_t)k;

    flags[idx] = (unsigned char)((pos ? 1 : 0) | (neg ? 2 : 0));
    cls_t[idx] = pos ? (float)s_lab[bm] : 0.0f;

    float4 t = make_float4(0.0f, 0.0f, 0.0f, 0.0f);
    if (pos) {
        const float4 mb = s_gt[bm];
        const float aw = a.z - a.x, ah = a.w - a.y;
        const float acx = a.x + 0.5f * aw, acy = a.y + 0.5f * ah;
        const float gw = mb.z - mb.x, gh = mb.w - mb.y;
        const float gcx = mb.x + 0.5f * gw, gcy = mb.y + 0.5f * gh;
        t.x = (gcx - acx) / aw;
        t.y = (gcy - acy) / ah;
        t.z = logf(gw / aw);
        t.w = logf(gh / ah);
    }
    ((float4*)reg_t)[idx] = t;
}

// ---------------------------------------------------------------------------
// Zero the coarse histograms (must run every call; graph replays accumulate).
// ---------------------------------------------------------------------------
__global__ void zero_hist_kernel(unsigned int* __restrict__ hist) {
    const size_t i = (size_t)blockIdx.x * blockDim.x + threadIdx.x;
    if (i < (size_t)NPROB * NBUCK) hist[i] = 0;
}

// ---------------------------------------------------------------------------
// Phase 1: 16-bit coarse histogram per (batch, pos|neg) problem.
// Inactive elements (NEG_INF filler) all hit one bucket -> block-aggregate in
// LDS and flush with a single global atomic per block to avoid contention.
// ---------------------------------------------------------------------------
__global__ void hist16_kernel(const float* __restrict__ score,
                              const unsigned char* __restrict__ flags,
                              unsigned int* __restrict__ hist) {
    __shared__ unsigned int inact[2];
    if (threadIdx.x < 2) inact[threadIdx.x] = 0;
    __syncthreads();

    const int n = blockIdx.y;
    const int k = blockIdx.x * blockDim.x + threadIdx.x;
    if (k < KANCH) {
        const size_t idx = (size_t)n * KANCH + (size_t)k;
        const unsigned char f = flags[idx];
        const unsigned int kb = orderKey(score[idx]) >> 16;
        if (f & 1) atomicAdd(&hist[(size_t)(n * 2 + 0) * NBUCK + kb], 1u);
        else       atomicAdd(&inact[0], 1u);
        if (f & 2) atomicAdd(&hist[(size_t)(n * 2 + 1) * NBUCK + kb], 1u);
        else       atomicAdd(&inact[1], 1u);
    }
    __syncthreads();
    if (threadIdx.x < 2) {
        const unsigned int c = inact[threadIdx.x];
        if (c) atomicAdd(&hist[(size_t)(n * 2 + threadIdx.x) * NBUCK +
                               (orderKey(NEG_INF_F) >> 16)], c);
    }
}

// ---------------------------------------------------------------------------
// Phase 2: suffix-scan the 64K buckets -> boundary bucket B and remainder r.
// If B is the NEG_INF bucket there are < 256 valid candidates: every valid
// candidate is selected, and NEG_INF fillers are masked out downstream anyway.
// ---------------------------------------------------------------------------
__global__ void scan16_kernel(const unsigned int* __restrict__ hist,
                              unsigned long long* __restrict__ kth,
                              unsigned int* __restrict__ bucketB,
                              unsigned int* __restrict__ wantR,
                              unsigned int* __restrict__ listCnt) {
    const int prob = blockIdx.x;
    const unsigned int* h = hist + (size_t)prob * NBUCK;

    __shared__ unsigned int chunkSum[256];
    const int t = threadIdx.x;
    const int base = NBUCK - (t + 1) * 256;   // chunk t = descending range
    unsigned int s = 0;
    for (int i = 0; i < 256; ++i) s += h[base + i];
    chunkSum[t] = s;
    __syncthreads();

    if (t == 0) {
        unsigned int cum = 0;
        int c = 255;
        for (int i = 0; i < 256; ++i) {       // highest buckets first
            const unsigned int cs = chunkSum[i];
            if (cum + cs >= NUM_SEL) { c = i; break; }
            cum += cs;
        }
        unsigned int B = 0;
        const int cbase = NBUCK - (c + 1) * 256;
        for (int b = cbase + 255; b >= cbase; --b) {
            const unsigned int cnt = h[b];
            if (cum + cnt >= NUM_SEL) { B = (unsigned int)b; break; }
            cum += cnt;
        }
        const unsigned int negInfBucket = orderKey(NEG_INF_F) >> 16;
        listCnt[prob] = 0;
        if (B == negInfBucket) {
            // fewer than NUM_SEL valid candidates: select all valid ones
            kth[prob]     = ((unsigned long long)(orderKey(NEG_INF_F) + 1u)) << 32;
            bucketB[prob] = 0xFFFFFFFFu;      // sentinel: skip tie resolution
            wantR[prob]   = 0;
        } else {
            bucketB[prob] = B;
            wantR[prob]   = NUM_SEL - cum;    // r >= 1, elements needed from B
            kth[prob]     = 0;
        }
    }
}

// ---------------------------------------------------------------------------
// Phase 3: compact boundary-bucket elements (tiny set) into a per-problem list.
// List order is nondeterministic but the k-th value derived from it is a set
// property -> final output stays deterministic.
// ---------------------------------------------------------------------------
__global__ void compact_kernel(const float* __restrict__ score,
                               const unsigned char* __restrict__ flags,
                               const unsigned int* __restrict__ bucketB,
                               unsigned int* __restrict__ listCnt,
                               unsigned long long* __restrict__ lists) {
    const int prob = blockIdx.y;              // 0..7
    const unsigned int B = bucketB[prob];
    if (B == 0xFFFFFFFFu) return;
    const int n = prob >> 1, sel = prob & 1;
    const int k = blockIdx.x * blockDim.x + threadIdx.x;
    if (k >= KANCH) return;

    const size_t idx = (size_t)n * KANCH + (size_t)k;
    const unsigned char f = flags[idx];
    const bool act = (sel == 0) ? ((f & 1) != 0) : ((f & 2) != 0);
    if (!act) return;                         // fillers live in the NEG_INF bucket
    const unsigned long long comp = compKey(score[idx], k);
    if ((unsigned int)(comp >> 48) != B) return;
    const unsigned int p = atomicAdd(&listCnt[prob], 1u);
    if (p < LISTCAP) lists[(size_t)prob * LISTCAP + p] = comp;
}

// ---------------------------------------------------------------------------
// Phase 4: exact r-th largest composite within the boundary bucket
// (radix select over the remaining 48 bits, 6 x 8-bit passes, tiny list).
// ---------------------------------------------------------------------------
__global__ void listselect_kernel(const unsigned long long* __restrict__ lists,
                                  const unsigned int* __restrict__ listCnt,
                                  const unsigned int* __restrict__ bucketB,
                                  const unsigned int* __restrict__ wantR,
                                  unsigned long long* __restrict__ kth) {
    const int prob = blockIdx.x;
    if (bucketB[prob] == 0xFFFFFFFFu) return; // kth already finalized

    __shared__ unsigned int       hist[256];
    __shared__ unsigned long long s_prefix, s_mask;
    __shared__ int                s_want;

    const int tid = threadIdx.x;
    const unsigned int cnt = min(listCnt[prob], (unsigned int)LISTCAP);
    const unsigned long long* L = lists + (size_t)prob * LISTCAP;

    if (tid == 0) {
        s_prefix = ((unsigned long long)bucketB[prob]) << 48;
        s_mask   = 0xFFFFull << 48;
        s_want   = (int)wantR[prob];
    }
    __syncthreads();

    for (int p = 5; p >= 0; --p) {
        const int shift = p * 8;
        hist[tid] = 0;                         // blockDim.x == 256
        __syncthreads();
        const unsigned long long prefix = s_prefix, mask = s_mask;
        for (unsigned int i = tid; i < cnt; i += blockDim.x) {
            const unsigned long long comp = L[i];
            if (((comp ^ prefix) & mask) == 0ull)
                atomicAdd(&hist[(unsigned int)((comp >> shift) & 0xFFull)], 1u);
        }
        __syncthreads();
        if (tid == 0) {
            unsigned int cum = 0;
            const int want = s_want;
            int chosen = 0;
            for (int b = 255; b >= 0; --b) {
                const unsigned int c = hist[b];
                if (cum + c >= (unsigned int)want) { chosen = b; break; }
                cum += c;
            }
            s_want   = want - (int)cum;
            s_prefix = prefix | ((unsigned long long)chosen << shift);
            s_mask   = mask   | (0xFFull << shift);
        }
        __syncthreads();
    }
    if (tid == 0) kth[prob] = s_prefix;        // exact 256-th largest composite
}

// ---------------------------------------------------------------------------
// Kernel C: apply subsample mask -> cls_weights / reg_weights.
// ---------------------------------------------------------------------------
__global__ void finalize_kernel(const float* __restrict__ score,
                                const unsigned char* __restrict__ flags,
                                const unsigned long long* __restrict__ kth,
                                float* __restrict__ cls_w,
                                float* __restrict__ reg_w) {
    const int n = blockIdx.y;
    const int k = blockIdx.x * blockDim.x + threadIdx.x;
    if (k >= KANCH) return;

    const size_t idx = (size_t)n * KANCH + (size_t)k;
    const unsigned char f = flags[idx];
    bool selp = false, seln = false;
    if (f & 3) {
        const unsigned long long comp = compKey(score[idx], k);
        if (f & 1) selp = comp >= kth[n * 2 + 0];
        if (f & 2) seln = comp >= kth[n * 2 + 1];
    }
    cls_w[idx] = (selp || seln) ? 1.0f : 0.0f; // mask subset of (pos|neg)
    reg_w[idx] = selp ? 1.0f : 0.0f;
}

// ---------------------------------------------------------------------------
extern "C" void kernel_launch(void* const* d_in, const int* in_sizes, int n_in,
                              void* d_out, int out_size, void* d_ws, size_t ws_size,
                              hipStream_t stream) {
    const float* anchors  = (const float*)d_in[0];   // (K, 4)
    const float* score    = (const float*)d_in[1];   // (N, K)
    const float* gt_boxes = (const float*)d_in[2];   // (N, M, 4)
    const int*   gt_lab   = (const int*)  d_in[3];   // (N, M)

    float* out   = (float*)d_out;
    float* cls_t = out;                                   // N*K
    float* reg_t = out + (size_t)NBATCH * KANCH;          // N*K*4
    float* cls_w = out + (size_t)NBATCH * KANCH * 5;      // N*K
    float* reg_w = out + (size_t)NBATCH * KANCH * 6;      // N*K

    // Workspace layout
    char* ws = (char*)d_ws;
    unsigned char*      flags   = (unsigned char*)ws;                    // N*K bytes
    size_t off = (size_t)NBATCH * KANCH;                                 // 960000
    unsigned long long* kth     = (unsigned long long*)(ws + off);       // 8 x u64
    off += NPROB * sizeof(unsigned long long);
    unsigned int*       bucketB = (unsigned int*)(ws + off);             // 8 x u32
    off += NPROB * sizeof(unsigned int);
    unsigned int*       wantR   = (unsigned int*)(ws + off);             // 8 x u32
    off += NPROB * sizeof(unsigned int);
    unsigned int*       listCnt = (unsigned int*)(ws + off);             // 8 x u32
    off += NPROB * sizeof(unsigned int);
    off = (off + 255) & ~(size_t)255;
    unsigned int*       hist    = (unsigned int*)(ws + off);             // 8*64K u32 = 2MB
    off += (size_t)NPROB * NBUCK * sizeof(unsigned int);
    unsigned long long* lists   = (unsigned long long*)(ws + off);       // 8*32K u64 = 2MB

    dim3 blk(256);
    dim3 gridK((KANCH + 255) / 256, NBATCH);
    dim3 gridP((KANCH + 255) / 256, NPROB);

    zero_hist_kernel<<<dim3((NPROB * NBUCK + 255) / 256), blk, 0, stream>>>(hist);
    anchor_target_kernel<<<gridK, blk, 0, stream>>>(anchors, gt_boxes, gt_lab,
                                                    cls_t, reg_t, flags);
    hist16_kernel<<<gridK, blk, 0, stream>>>(score, flags, hist);
    scan16_kernel<<<dim3(NPROB), blk, 0, stream>>>(hist, kth, bucketB, wantR, listCnt);
    compact_kernel<<<gridP, blk, 0, stream>>>(score, flags, bucketB, listCnt, lists);
    listselect_kernel<<<dim3(NPROB), blk, 0, stream>>>(lists, listCnt, bucketB,
                                                       wantR, kth);
    finalize_kernel<<<gridK, blk, 0, stream>>>(score, flags, kth, cls_w, reg_w);
}